// EncGraphConv_249108103973
// MI455X (gfx1250) — compile-verified
//
#include <hip/hip_runtime.h>

typedef __attribute__((ext_vector_type(2))) float v2f;
typedef __attribute__((ext_vector_type(8))) float v8f;

// Problem dimensions (fixed by the reference).
constexpr int N_NODES = 4096;
constexpr int BT      = 192;
constexpr int D_IN    = 32;
constexpr int M_MATS  = 5;
constexpr int D_OUT   = 64;
constexpr int F       = BT * D_IN;          // 6144 features per node
constexpr int KDIM    = M_MATS * D_IN;      // 160 = GEMM K
constexpr int LDA     = 162;                // LDS pitch for A tile (breaks 160%64 bank pattern)

// ---------------------------------------------------------------------------
// Kernel 1: transpose x [F, N] -> x0 [N, F]  (hop 0, node-major)
// ---------------------------------------------------------------------------
__global__ void transpose_kernel(const float* __restrict__ in, float* __restrict__ out) {
    __shared__ float tile[32][33];
    const int n0 = blockIdx.x * 32;
    const int f0 = blockIdx.y * 32;
#pragma unroll
    for (int i = 0; i < 4; ++i) {
        const int f = f0 + threadIdx.y + i * 8;
        tile[threadIdx.y + i * 8][threadIdx.x] = in[(size_t)f * N_NODES + n0 + threadIdx.x];
    }
    __syncthreads();
#pragma unroll
    for (int i = 0; i < 4; ++i) {
        const int n = n0 + threadIdx.y + i * 8;
        out[(size_t)n * F + f0 + threadIdx.x] = tile[threadIdx.x][threadIdx.y + i * 8];
    }
}

// ---------------------------------------------------------------------------
// Kernel 2: zero-fill (hops 1..4 accumulate via atomics)
// ---------------------------------------------------------------------------
__global__ void zero_kernel(float4* __restrict__ p, long long n4) {
    long long i = (long long)blockIdx.x * blockDim.x + threadIdx.x;
    const long long stride = (long long)gridDim.x * blockDim.x;
    const float4 z = make_float4(0.f, 0.f, 0.f, 0.f);
    for (; i < n4; i += stride) p[i] = z;
}

// ---------------------------------------------------------------------------
// Kernel 3: SPMM  y[r,:] += v * x[c,:]   (node-major, one block per edge)
// ---------------------------------------------------------------------------
__global__ void spmm_kernel(const int* __restrict__ rows, const int* __restrict__ cols,
                            const float* __restrict__ vals,
                            const float* __restrict__ xin, float* __restrict__ yout) {
    const int e = blockIdx.x;
    const int r = rows[e];
    const int c = cols[e];
    const float v = vals[e];
    const float4* __restrict__ src = (const float4*)(xin + (size_t)c * F);
    float* __restrict__ dst = yout + (size_t)r * F;
#pragma unroll
    for (int i = threadIdx.x; i < F / 4; i += 256) {
        const float4 t = src[i];
        atomicAdd(dst + 4 * i + 0, v * t.x);
        atomicAdd(dst + 4 * i + 1, v * t.y);
        atomicAdd(dst + 4 * i + 2, v * t.z);
        atomicAdd(dst + 4 * i + 3, v * t.w);
    }
}

// ---------------------------------------------------------------------------
// Kernel 4: WMMA GEMM  out[bt*N+n, :] = xs_row @ W + bias
//   xs_row[m*32+d] = hops[m][n*F + bt*32 + d]
//   128 threads = 4 wave32; block covers 8 row-tiles of 16 rows x 64 cols.
// ---------------------------------------------------------------------------
constexpr int TILES_PER_BLOCK = 8;

__global__ void
__launch_bounds__(128)
gemm_wmma_kernel(const float* __restrict__ hops,   // 5 x [N, F]
                 const float* __restrict__ weight, // [160, 64]
                 const float* __restrict__ bias,   // [64]
                 float* __restrict__ out) {        // [BT*N, 64]
    __shared__ float w_lds[KDIM * D_OUT];    // 40 KB
    __shared__ float a_lds[16 * LDA];        // ~10.1 KB

    const int tid = threadIdx.x;
    // Stage the full weight once per block.
    for (int i = tid; i < KDIM * D_OUT; i += 128) w_lds[i] = weight[i];

    const int lane = tid & 31;
    const int wid  = tid >> 5;      // 0..3 -> 16-column slice
    const int hi   = lane >> 4;     // lane group (0: lanes 0-15, 1: lanes 16-31)
    const int ln   = lane & 15;
    const int col0 = wid * 16;
    const size_t hop_elems = (size_t)N_NODES * F;
    const float bval = bias[col0 + ln];

    for (int t = 0; t < TILES_PER_BLOCK; ++t) {
        const int tile_id = blockIdx.x * TILES_PER_BLOCK + t;
        const int row0 = tile_id * 16;             // global row in [0, BT*N)
        const int bt   = row0 / N_NODES;           // tiles never straddle bt (N%16==0)
        const int n0   = row0 % N_NODES;

        __syncthreads();  // protect a_lds from previous iteration's readers

        // Cooperative gather of the 16x160 A tile into LDS.
        for (int i = tid; i < 16 * KDIM; i += 128) {
            const int rrow = i / KDIM;
            const int cc   = i % KDIM;
            const int m    = cc >> 5;              // which hop
            const int d    = cc & 31;              // feature within D_IN
            a_lds[rrow * LDA + cc] =
                hops[(size_t)m * hop_elems + (size_t)(n0 + rrow) * F + bt * D_IN + d];
        }
        __syncthreads();

        v8f acc = {};
#pragma unroll
        for (int k0 = 0; k0 < KDIM; k0 += 4) {
            const int kA = k0 + 2 * hi;
            // A fragment: row = ln, K = kA, kA+1 (contiguous in LDS, 8B aligned)
            const v2f a = *(const v2f*)&a_lds[ln * LDA + kA];
            // B fragment: col = col0+ln, K = kA, kA+1
            v2f b;
            b.x = w_lds[(kA + 0) * D_OUT + col0 + ln];
            b.y = w_lds[(kA + 1) * D_OUT + col0 + ln];
            acc = __builtin_amdgcn_wmma_f32_16x16x4_f32(
                /*neg_a=*/false, a, /*neg_b=*/false, b,
                /*c_mod=*/(short)0, acc, /*reuse_a=*/false, /*reuse_b=*/false);
        }

        // D layout: VGPR v -> row (v + 8*hi), col ln.
#pragma unroll
        for (int v = 0; v < 8; ++v) {
            const int grow = row0 + v + 8 * hi;
            out[(size_t)grow * D_OUT + col0 + ln] = acc[v] + bval;
        }
    }
}

// ---------------------------------------------------------------------------
// Host-side launch
// ---------------------------------------------------------------------------
extern "C" void kernel_launch(void* const* d_in, const int* in_sizes, int n_in,
                              void* d_out, int out_size, void* d_ws, size_t ws_size,
                              hipStream_t stream) {
    const float* x       = (const float*)d_in[0];
    const int*   s0_rows = (const int*)d_in[1];
    const int*   s0_cols = (const int*)d_in[2];
    const float* s0_vals = (const float*)d_in[3];
    const int*   s1_rows = (const int*)d_in[4];
    const int*   s1_cols = (const int*)d_in[5];
    const float* s1_vals = (const float*)d_in[6];
    const float* weight  = (const float*)d_in[7];
    const float* biases  = (const float*)d_in[8];
    float* out = (float*)d_out;

    const size_t hop_elems = (size_t)N_NODES * F;   // 25,165,824 floats each
    float* h0 = (float*)d_ws;                       // x0 (transposed x)
    float* h1 = h0 + hop_elems;                     // A0 x0
    float* h2 = h1 + hop_elems;                     // A0^2 x0
    float* h3 = h2 + hop_elems;                     // A1 x0
    float* h4 = h3 + hop_elems;                     // A1^2 x0

    // 1) x0 = x.T  (node-major)
    transpose_kernel<<<dim3(N_NODES / 32, F / 32), dim3(32, 8), 0, stream>>>(x, h0);

    // 2) zero hops 1..4
    zero_kernel<<<4096, 256, 0, stream>>>((float4*)h1, (long long)(4 * hop_elems / 4));

    // 3) diffusion hops
    constexpr int E_EDGES = 65536;
    spmm_kernel<<<E_EDGES, 256, 0, stream>>>(s0_rows, s0_cols, s0_vals, h0, h1);
    spmm_kernel<<<E_EDGES, 256, 0, stream>>>(s0_rows, s0_cols, s0_vals, h1, h2);
    spmm_kernel<<<E_EDGES, 256, 0, stream>>>(s1_rows, s1_cols, s1_vals, h0, h3);
    spmm_kernel<<<E_EDGES, 256, 0, stream>>>(s1_rows, s1_cols, s1_vals, h3, h4);

    // 4) fused gather + WMMA GEMM + bias
    const int n_tiles = (BT * N_NODES) / 16;        // 49152
    gemm_wmma_kernel<<<n_tiles / TILES_PER_BLOCK, 128, 0, stream>>>(h0, weight, biases, out);
}